// LSConv_32023276159247
// MI455X (gfx1250) — compile-verified
//
#include <hip/hip_runtime.h>
#include <math.h>

// LSConv for MI455X (gfx1250). fp32 throughout; GEMMs on V_WMMA_F32_16X16X4_F32.
// Scratch layout (floats): z1[4.19M] | z2[4.19M] | z4[9.44M] | stats[2048]  ~= 71.3 MB.

#define HW 1024     // 32*32
#define EPSV 1e-5f

typedef __attribute__((ext_vector_type(2))) float v2f;
typedef __attribute__((ext_vector_type(8))) float v8f;

// ---------------------------------------------------------------------------
// GEMM: Y[b, o, hw] = sum_k Wt[o,k] * X[b, k, hw]  (+ fused epilogue)
//   Wt: (M, K) row-major; X: (16, K, 1024); Y: (16, M, 1024)
//   relu_mode=1: y = relu((v + cb - bm)*g*rsqrt(bv+eps) + bb)   (conv bias + BN + ReLU)
//   relu_mode=0: y = v + cb                                      (bias only, for cv4)
// One wave -> 16(M) x 64(N) tile; 4 f32 accumulators; A fragment reused 4x.
// f32 WMMA fragment layout (ISA 7.12.2):
//   A 16x4:  lane m=lane&15; VGPR v holds A[m, k + 2*(lane>>4) + v]
//   B 4x16:  lane n=lane&15; VGPR v holds B[k + 2*(lane>>4) + v, n]
//   C 16x16: VGPR r holds D[r + 8*(lane>>4), lane&15]
// ---------------------------------------------------------------------------
__global__ void gemm_wmma_f32(const float* __restrict__ Wt,
                              const float* __restrict__ X,
                              float* __restrict__ Y,
                              const float* __restrict__ cb,
                              const float* __restrict__ bg,
                              const float* __restrict__ bb,
                              const float* __restrict__ bm,
                              const float* __restrict__ bv,
                              int M, int K, int Mtiles, int relu_mode)
{
    const int lane = threadIdx.x & 31;
    const int wid  = threadIdx.x >> 5;
    const int gw   = blockIdx.x * 8 + wid;     // 8 waves / block
    const int mt   = gw % Mtiles;
    const int nt   = gw / Mtiles;              // 256 column super-tiles of width 64
    const int m0   = mt * 16;
    const int ncol = nt * 64;
    const int b    = ncol >> 10;               // 1024 pixels per image, tiles never straddle
    const int hw0  = ncol & 1023;

    const int lm = lane & 15;                  // M row (A) / N col (B)
    const int lk = (lane >> 4) << 1;           // K-pair select {0,2}

    const float* __restrict__ Xb = X + (size_t)b * K * HW + hw0;
    const float* __restrict__ Ar = Wt + (size_t)(m0 + lm) * K + lk;

    v8f zero = {0.f, 0.f, 0.f, 0.f, 0.f, 0.f, 0.f, 0.f};
    v8f acc[4];
#pragma unroll
    for (int s = 0; s < 4; ++s) acc[s] = zero;

    for (int k = 0; k < K; k += 4) {
        v2f a = *(const v2f*)(Ar + k);                         // A[m, k+lk], A[m, k+lk+1]
        const float* Bk = Xb + (size_t)(k + lk) * HW + lm;
#pragma unroll
        for (int s = 0; s < 4; ++s) {
            v2f bf;
            bf.x = Bk[s * 16];                                 // B[k+lk,   n]
            bf.y = Bk[s * 16 + HW];                            // B[k+lk+1, n]
            acc[s] = __builtin_amdgcn_wmma_f32_16x16x4_f32(
                false, a, false, bf, (short)0, acc[s], false, false);
        }
    }

    const int hi = (lane >> 4) << 3;                           // +8 rows for upper half-wave
    float osc[8], obi[8];
#pragma unroll
    for (int r = 0; r < 8; ++r) {
        int o = m0 + r + hi;
        if (relu_mode) {
            float sc = bg[o] * rsqrtf(bv[o] + EPSV);
            osc[r] = sc;
            obi[r] = (cb[o] - bm[o]) * sc + bb[o];
        } else {
            osc[r] = 1.f;
            obi[r] = cb[o];
        }
    }
#pragma unroll
    for (int s = 0; s < 4; ++s) {
        int ncl = hw0 + s * 16 + lm;
#pragma unroll
        for (int r = 0; r < 8; ++r) {
            int o = m0 + r + hi;
            float val = acc[s][r] * osc[r] + obi[r];
            if (relu_mode) val = fmaxf(val, 0.f);
            Y[(size_t)b * M * HW + (size_t)o * HW + ncl] = val;
        }
    }
}

// ---------------------------------------------------------------------------
// Depthwise 7x7 (pad 3) + BN.  z1,z2: (16, 256, 32, 32). L2-resident gather.
// ---------------------------------------------------------------------------
__global__ void dwconv7_bn(const float* __restrict__ z1,
                           const float* __restrict__ wt,   // (256,1,7,7)
                           const float* __restrict__ cb,
                           const float* __restrict__ bg,
                           const float* __restrict__ bb,
                           const float* __restrict__ bm,
                           const float* __restrict__ bv,
                           float* __restrict__ z2)
{
    int idx = blockIdx.x * blockDim.x + threadIdx.x;   // 16*256*1024
    int hw = idx & 1023;
    int ch = (idx >> 10) & 255;
    int b  = idx >> 18;
    int h = hw >> 5, w = hw & 31;
    const float* src = z1 + ((size_t)(b * 256 + ch)) * HW;
    const float* wk  = wt + ch * 49;
    float acc = 0.f;
#pragma unroll
    for (int i = 0; i < 7; ++i) {
        int yy = h + i - 3;
        if ((unsigned)yy >= 32u) continue;
#pragma unroll
        for (int j = 0; j < 7; ++j) {
            int xx = w + j - 3;
            if ((unsigned)xx < 32u) acc += wk[i * 7 + j] * src[yy * 32 + xx];
        }
    }
    acc += cb[ch];
    float sc = bg[ch] * rsqrtf(bv[ch] + EPSV);
    z2[idx] = (acc - bm[ch]) * sc + bb[ch];
}

// ---------------------------------------------------------------------------
// GroupNorm stats. Group (b, cc) = 9 channels * 1024 px = 9216 CONTIGUOUS floats
// (since 576 = 64*9 and layout is (b,576,1024)). 1 block / group.
// ---------------------------------------------------------------------------
__global__ void gn_stats(const float* __restrict__ z4, float* __restrict__ stats)
{
    __shared__ float s1[256], s2[256];
    int gid = blockIdx.x;                       // b*64 + cc, 1024 total
    int tid = threadIdx.x;
    const float* base = z4 + (size_t)gid * 9216;
    float a = 0.f, q = 0.f;
    for (int i = tid; i < 9216; i += 256) { float t = base[i]; a += t; q += t * t; }
    s1[tid] = a; s2[tid] = q;
    __syncthreads();
    for (int off = 128; off > 0; off >>= 1) {
        if (tid < off) { s1[tid] += s1[tid + off]; s2[tid] += s2[tid + off]; }
        __syncthreads();
    }
    if (tid == 0) {
        float mean = s1[0] * (1.f / 9216.f);
        float var  = s2[0] * (1.f / 9216.f) - mean * mean;
        stats[gid * 2]     = mean;
        stats[gid * 2 + 1] = rsqrtf(var + EPSV);
    }
}

// ---------------------------------------------------------------------------
// SKA: per (b, cc, hw): GN-normalize 9 logits, softmax, then for 8 channels
// c = g*64+cc aggregate 3x3 zero-padded neighborhood; output BN + residual.
// ---------------------------------------------------------------------------
__global__ void ska_kernel(const float* __restrict__ x,
                           const float* __restrict__ z4,
                           const float* __restrict__ stats,
                           const float* __restrict__ gg,
                           const float* __restrict__ gb,
                           const float* __restrict__ og,
                           const float* __restrict__ ob,
                           const float* __restrict__ om,
                           const float* __restrict__ ov,
                           float* __restrict__ out)
{
    int idx = blockIdx.x * blockDim.x + threadIdx.x;   // 16*64*1024
    int hw = idx & 1023;
    int cc = (idx >> 10) & 63;
    int b  = idx >> 16;
    int h = hw >> 5, w = hw & 31;

    float mu = stats[(b * 64 + cc) * 2];
    float rs = stats[(b * 64 + cc) * 2 + 1];

    const float* zp = z4 + ((size_t)(b * 576 + cc * 9)) * HW + hw;
    float wv[9];
    float mx = -1e30f;
#pragma unroll
    for (int k = 0; k < 9; ++k) {
        float t = (zp[(size_t)k * HW] - mu) * rs * gg[cc * 9 + k] + gb[cc * 9 + k];
        wv[k] = t;
        mx = fmaxf(mx, t);
    }
    float ssum = 0.f;
#pragma unroll
    for (int k = 0; k < 9; ++k) { float e = __expf(wv[k] - mx); wv[k] = e; ssum += e; }
    float inv = 1.f / ssum;
#pragma unroll
    for (int k = 0; k < 9; ++k) wv[k] *= inv;

#pragma unroll
    for (int g = 0; g < 8; ++g) {
        int c = g * 64 + cc;
        const float* xs = x + ((size_t)(b * 512 + c)) * HW;
        float acc = 0.f;
        int k = 0;
#pragma unroll
        for (int i = 0; i < 3; ++i) {
            int yy = h + i - 1;
#pragma unroll
            for (int j = 0; j < 3; ++j, ++k) {
                int xx = w + j - 1;
                if ((unsigned)yy < 32u && (unsigned)xx < 32u)
                    acc += wv[k] * xs[yy * 32 + xx];
            }
        }
        float sc = og[c] * rsqrtf(ov[c] + EPSV);
        out[((size_t)(b * 512 + c)) * HW + hw] = (acc - om[c]) * sc + ob[c] + xs[hw];
    }
}

// ---------------------------------------------------------------------------
extern "C" void kernel_launch(void* const* d_in, const int* in_sizes, int n_in,
                              void* d_out, int out_size, void* d_ws, size_t ws_size,
                              hipStream_t stream)
{
    const float* x     = (const float*)d_in[0];
    const float* cv1_w = (const float*)d_in[1];
    const float* cv1_b = (const float*)d_in[2];
    const float* bn1_g = (const float*)d_in[3];
    const float* bn1_b = (const float*)d_in[4];
    const float* bn1_m = (const float*)d_in[5];
    const float* bn1_v = (const float*)d_in[6];
    const float* cv2_w = (const float*)d_in[7];
    const float* cv2_b = (const float*)d_in[8];
    const float* bn2_g = (const float*)d_in[9];
    const float* bn2_b = (const float*)d_in[10];
    const float* bn2_m = (const float*)d_in[11];
    const float* bn2_v = (const float*)d_in[12];
    const float* cv3_w = (const float*)d_in[13];
    const float* cv3_b = (const float*)d_in[14];
    const float* bn3_g = (const float*)d_in[15];
    const float* bn3_b = (const float*)d_in[16];
    const float* bn3_m = (const float*)d_in[17];
    const float* bn3_v = (const float*)d_in[18];
    const float* cv4_w = (const float*)d_in[19];
    const float* cv4_b = (const float*)d_in[20];
    const float* gn_g  = (const float*)d_in[21];
    const float* gn_b  = (const float*)d_in[22];
    const float* bno_g = (const float*)d_in[23];
    const float* bno_b = (const float*)d_in[24];
    const float* bno_m = (const float*)d_in[25];
    const float* bno_v = (const float*)d_in[26];

    float* ws  = (float*)d_ws;
    float* Z1  = ws;                                  // 16*256*1024 = 4,194,304 f
    float* Z2  = ws + (size_t)4194304;                // 4,194,304 f
    float* Z4  = ws + (size_t)8388608;                // 16*576*1024 = 9,437,184 f
    float* ST  = ws + (size_t)(8388608 + 9437184);    // 2048 f
    float* out = (float*)d_out;

    // cv1: 256x512 GEMM over 16384 px, BN+ReLU.  waves = 16*256 -> 512 blocks.
    gemm_wmma_f32<<<512, 256, 0, stream>>>(cv1_w, x, Z1,
                                           cv1_b, bn1_g, bn1_b, bn1_m, bn1_v,
                                           256, 512, 16, 1);
    // depthwise 7x7 + BN2
    dwconv7_bn<<<16384, 256, 0, stream>>>(Z1, cv2_w, cv2_b,
                                          bn2_g, bn2_b, bn2_m, bn2_v, Z2);
    // cv3: 256x256 GEMM, BN+ReLU. Writes into Z1 (z1 dead after dwconv).
    gemm_wmma_f32<<<512, 256, 0, stream>>>(cv3_w, Z2, Z1,
                                           cv3_b, bn3_g, bn3_b, bn3_m, bn3_v,
                                           256, 256, 16, 1);
    // cv4: 576x256 GEMM, bias only. waves = 36*256 -> 1152 blocks.
    gemm_wmma_f32<<<1152, 256, 0, stream>>>(cv4_w, Z1, Z4,
                                            cv4_b, cv4_b, cv4_b, cv4_b, cv4_b,
                                            576, 256, 36, 0);
    // GroupNorm stats: 16*64 groups
    gn_stats<<<1024, 256, 0, stream>>>(Z4, ST);
    // SKA + output BN + residual
    ska_kernel<<<4096, 256, 0, stream>>>(x, Z4, ST, gn_g, gn_b,
                                         bno_g, bno_b, bno_m, bno_v, out);
}